// IRazorEmbedding_70282844831820
// MI455X (gfx1250) — compile-verified
//
#include <hip/hip_runtime.h>
#include <math.h>

typedef __attribute__((ext_vector_type(2))) float v2f;
typedef __attribute__((ext_vector_type(8))) float v8f;

#define BATCH   8192
#define FIELDS  32
#define DIM     32
#define REGIONS 6
#define NPERF   (BATCH * DIM)     // elements per field for BN stats
#define EPSV    1e-5f

// workspace layout (floats)
#define WS_SUM   0                 // [32] per-field sum
#define WS_SQ    32                // [32] per-field sum of squares
#define WS_SCALE 64                // [32][32] A[f][d] = rstd[f]*coef[f][d]
#define WS_BIAS  (64 + 1024)      // [32][32] B[f][d] = -mean[f]*A[f][d]

__global__ void irazor_zero_kernel(float* __restrict__ ws) {
    if (threadIdx.x < 64) ws[threadIdx.x] = 0.0f;
}

// Pass 1: per-field sum / sumsq. Blocks are partitioned by field so each lane
// accumulates in registers; one shuffle-tree + 2 global float atomics per wave.
__global__ void irazor_stats_kernel(const int* __restrict__ ids,
                                    const float4* __restrict__ tab4,
                                    float* __restrict__ ws) {
    const int BPF   = 16;                       // blocks per field
    const int f     = blockIdx.x / BPF;
    const int blkIF = blockIdx.x % BPF;
    const int lane  = threadIdx.x & 31;
    const int wave  = threadIdx.x >> 5;
    const int gw    = blkIF * (blockDim.x >> 5) + wave;   // wave index within field
    const int nw    = BPF * (blockDim.x >> 5);            // waves per field
    const int sub   = lane >> 3;                // which of 4 batch rows
    const int q     = lane & 7;                 // float4 slot within a 32-f32 row

    float s = 0.0f, s2 = 0.0f;
    for (int c = gw; c < BATCH / 4; c += nw) {
        int b  = c * 4 + sub;
        int id = ids[b * FIELDS + f];           // 8-lane broadcast load
        float4 x = tab4[(size_t)id * 8 + q];    // coalesced 128B line per row
        s  += x.x + x.y + x.z + x.w;
        s2 += x.x * x.x + x.y * x.y + x.z * x.z + x.w * x.w;
    }
#pragma unroll
    for (int off = 16; off; off >>= 1) {
        s  += __shfl_xor(s,  off, 32);
        s2 += __shfl_xor(s2, off, 32);
    }
    if (lane == 0) {
        atomicAdd(&ws[WS_SUM + f], s);
        atomicAdd(&ws[WS_SQ  + f], s2);
    }
}

// Single-wave prep: mean/rstd per field, softmax over regions, then the
// 32x6 @ 6x32 einsum via 8x V_WMMA_F32_16X16X4_F32 (K padded 6->8, 2 k-steps,
// 2x2 MN tiles). Emits fused affine tables A,B.
__global__ void irazor_prep_kernel(const float* __restrict__ frw,    // [32][6]
                                   const float* __restrict__ rmask,  // [6][32]
                                   float* __restrict__ ws) {
    const int l     = threadIdx.x;      // 0..31, exactly one wave, no divergence
    const int half  = l >> 4;
    const int lanem = l & 15;
    const float invN = 1.0f / (float)NPERF;

    float meanv = ws[WS_SUM + l] * invN;
    float varv  = ws[WS_SQ  + l] * invN - meanv * meanv;
    float rstdv = rsqrtf(varv + EPSV);

    const int ka = 2 * half;            // K pair base within a k-step

#pragma unroll
    for (int mt = 0; mt < 2; ++mt) {
        // softmax for this lane's A-row field
        int mrow = mt * 16 + lanem;
        float wv[8];
        float mx = -1e30f;
#pragma unroll
        for (int r = 0; r < REGIONS; ++r) { wv[r] = frw[mrow * REGIONS + r]; mx = fmaxf(mx, wv[r]); }
        float sum = 0.0f;
#pragma unroll
        for (int r = 0; r < REGIONS; ++r) { wv[r] = __expf(wv[r] - mx); sum += wv[r]; }
        float inv = 1.0f / sum;
#pragma unroll
        for (int r = 0; r < 8; ++r) wv[r] = (r < REGIONS) ? wv[r] * inv : 0.0f;

        // A operands: ISA A-layout 16x4 f32: lane = M&15, VGPR k = 2*half+{0,1}
        v2f a0, a1;
        a0.x = wv[ka];     a0.y = wv[ka + 1];
        a1.x = wv[4 + ka]; a1.y = wv[5 + ka];

#pragma unroll
        for (int nt = 0; nt < 2; ++nt) {
            int ncol = nt * 16 + lanem;
            float bm[8];
#pragma unroll
            for (int r = 0; r < 8; ++r) bm[r] = (r < REGIONS) ? rmask[r * DIM + ncol] : 0.0f;
            v2f b0, b1;
            b0.x = bm[ka];     b0.y = bm[ka + 1];
            b1.x = bm[4 + ka]; b1.y = bm[5 + ka];

            v8f c = {};
            c = __builtin_amdgcn_wmma_f32_16x16x4_f32(false, a0, false, b0,
                                                      (short)0, c, false, false);
            c = __builtin_amdgcn_wmma_f32_16x16x4_f32(false, a1, false, b1,
                                                      (short)0, c, false, false);
#pragma unroll
            for (int v = 0; v < 8; ++v) {
                int M = mt * 16 + v + half * 8;     // D-layout: VGPR v, lane-half
                float rs = __shfl(rstdv, M, 32);
                float mn = __shfl(meanv, M, 32);
                float A  = rs * c[v];
                ws[WS_SCALE + M * DIM + ncol] = A;
                ws[WS_BIAS  + M * DIM + ncol] = -mn * A;
            }
        }
    }
}

// Pass 2: re-gather (L2-hot: table is 25.6MB << 192MB L2), one FMA per element.
__global__ void irazor_out_kernel(const int* __restrict__ ids,
                                  const float4* __restrict__ tab4,
                                  const float* __restrict__ ws,
                                  float4* __restrict__ out4) {
    const int lane = threadIdx.x & 31;
    const int gw   = blockIdx.x * (blockDim.x >> 5) + (threadIdx.x >> 5);
    const int nw   = gridDim.x * (blockDim.x >> 5);
    const int sub  = lane >> 3;
    const int q    = lane & 7;
    const int CH   = (BATCH * FIELDS) / 4;

    for (int c = gw; c < CH; c += nw) {
        int row = c * 4 + sub;
        int f   = row & (FIELDS - 1);
        int id  = ids[row];
        float4 x  = tab4[(size_t)id * 8 + q];
        float4 A  = *(const float4*)&ws[WS_SCALE + f * DIM + q * 4];
        float4 Bb = *(const float4*)&ws[WS_BIAS  + f * DIM + q * 4];
        float4 o;
        o.x = fmaf(x.x, A.x, Bb.x);
        o.y = fmaf(x.y, A.y, Bb.y);
        o.z = fmaf(x.z, A.z, Bb.z);
        o.w = fmaf(x.w, A.w, Bb.w);
        out4[(size_t)row * 8 + q] = o;
    }
}

extern "C" void kernel_launch(void* const* d_in, const int* in_sizes, int n_in,
                              void* d_out, int out_size, void* d_ws, size_t ws_size,
                              hipStream_t stream) {
    const int*   ids   = (const int*)d_in[0];      // input_ids [8192][32]
    const float* tab   = (const float*)d_in[1];    // emb_table [200000][32]
    const float* frw   = (const float*)d_in[2];    // field_region_weights [32][6][1]
    const float* rmask = (const float*)d_in[3];    // region_mask [6][32]
    float* ws  = (float*)d_ws;
    float* out = (float*)d_out;

    irazor_zero_kernel<<<1, 64, 0, stream>>>(ws);
    irazor_stats_kernel<<<FIELDS * 16, 256, 0, stream>>>(ids, (const float4*)tab, ws);
    irazor_prep_kernel<<<1, 32, 0, stream>>>(frw, rmask, ws);
    irazor_out_kernel<<<1024, 256, 0, stream>>>(ids, (const float4*)tab, ws,
                                                (float4*)out);
}